// Time_Mix_10376640987681
// MI455X (gfx1250) — compile-verified
//
#include <hip/hip_runtime.h>
#include <hip/hip_bf16.h>
#include <math.h>

#define BB 2
#define TT 1024
#define EE 1024
#define HH 16
#define DD 64
#define RR 32
#define BT (BB*TT)

typedef float v2f __attribute__((ext_vector_type(2)));
typedef float v8f __attribute__((ext_vector_type(8)));

// ---- 16x16 f32 tile GEMM: C = A(16xK, row-major lda) * B(Kx16, row-major ldb)
// Lane layout per CDNA5 ISA 7.12.2:
//  A: lanes 0-15 -> M=lane, VGPR0/1 = K=k0,k0+1 ; lanes 16-31 -> K=k0+2,k0+3
//  B: mirrored with N on lanes
//  C: lanes 0-15 col=lane rows 0-7 in c[0..7]; lanes 16-31 col=lane-16 rows 8-15
__device__ __forceinline__ v8f wmma_tile_f32(const float* A, int lda,
                                             const float* B, int ldb, int K) {
  const int lane = threadIdx.x & 31;
  const int lo   = lane & 15;
  const int kh   = (lane >> 4) << 1;  // 0 or 2
  v8f c = {};
#if __has_builtin(__builtin_amdgcn_wmma_f32_16x16x4_f32)
  for (int kk = 0; kk < K; kk += 4) {
    v2f a, b;
    a.x = A[(size_t)lo * lda + (kk + kh)];
    a.y = A[(size_t)lo * lda + (kk + kh + 1)];
    b.x = B[(size_t)(kk + kh) * ldb + lo];
    b.y = B[(size_t)(kk + kh + 1) * ldb + lo];
    c = __builtin_amdgcn_wmma_f32_16x16x4_f32(false, a, false, b,
                                              (short)0, c, false, false);
  }
#else
  for (int i = 0; i < 8; ++i) {
    int row = (lane < 16) ? i : i + 8;
    float acc = 0.f;
    for (int k = 0; k < K; ++k) acc += A[(size_t)row * lda + k] * B[(size_t)k * ldb + lo];
    c[i] = acc;
  }
#endif
  return c;
}

// ---- 2x2 register-blocked variant: 32x32 output per wave, fragments shared.
__device__ __forceinline__ void wmma_tile_f32_2x2(const float* A, int lda,
                                                  const float* B, int ldb, int K,
                                                  v8f c[2][2]) {
  const int lane = threadIdx.x & 31;
  const int lo   = lane & 15;
  const int kh   = (lane >> 4) << 1;
#if __has_builtin(__builtin_amdgcn_wmma_f32_16x16x4_f32)
  for (int kk = 0; kk < K; kk += 4) {
    v2f a0, a1, b0, b1;
    a0.x = A[(size_t)lo * lda + (kk + kh)];
    a0.y = A[(size_t)lo * lda + (kk + kh + 1)];
    a1.x = A[(size_t)(lo + 16) * lda + (kk + kh)];
    a1.y = A[(size_t)(lo + 16) * lda + (kk + kh + 1)];
    b0.x = B[(size_t)(kk + kh) * ldb + lo];
    b0.y = B[(size_t)(kk + kh + 1) * ldb + lo];
    b1.x = B[(size_t)(kk + kh) * ldb + lo + 16];
    b1.y = B[(size_t)(kk + kh + 1) * ldb + lo + 16];
    c[0][0] = __builtin_amdgcn_wmma_f32_16x16x4_f32(false, a0, false, b0, (short)0, c[0][0], false, false);
    c[0][1] = __builtin_amdgcn_wmma_f32_16x16x4_f32(false, a0, false, b1, (short)0, c[0][1], false, false);
    c[1][0] = __builtin_amdgcn_wmma_f32_16x16x4_f32(false, a1, false, b0, (short)0, c[1][0], false, false);
    c[1][1] = __builtin_amdgcn_wmma_f32_16x16x4_f32(false, a1, false, b1, (short)0, c[1][1], false, false);
  }
#else
  for (int mi = 0; mi < 2; ++mi)
    for (int nj = 0; nj < 2; ++nj)
      for (int i = 0; i < 8; ++i) {
        int row = ((lane < 16) ? i : i + 8) + mi * 16;
        float acc = 0.f;
        for (int k = 0; k < K; ++k)
          acc += A[(size_t)row * lda + k] * B[(size_t)k * ldb + nj * 16 + lo];
        c[mi][nj][i] = acc;
      }
#endif
}

// ---------------- K1: token shift, diff, z ----------------
__global__ void k_prep(const float* __restrict__ x, const float* __restrict__ mu_x,
                       float* __restrict__ z, float* __restrict__ diff) {
  int idx = blockIdx.x * blockDim.x + threadIdx.x;
  if (idx >= BT * EE) return;
  int e = idx & (EE - 1);
  int m = idx >> 10;           // EE == 1024
  int t = m & (TT - 1);
  float xv = x[idx];
  float lx = (t > 0) ? x[idx - EE] : 0.f;
  float df = lx - xv;
  float mut = xv + df * mu_x[e];
  z[idx]    = xv + df * mut;
  diff[idx] = df;
}

// ---------------- K2a: hdn = tanh(z @ lora_A_w + b) ----------------
__global__ void k_loraA(const float* __restrict__ z, const float* __restrict__ Aw,
                        const float* __restrict__ Ab, float* __restrict__ hdn) {
  int mt = blockIdx.x, nt = blockIdx.y;
  int lane = threadIdx.x & 31, lo = lane & 15;
  int g = nt >> 1, r0 = (nt & 1) * 16;          // 16-col tile stays within one branch
  const float* Ap = z + (size_t)mt * 16 * EE;
  const float* Bp = Aw + (size_t)g * EE * RR + r0;
  v8f c = wmma_tile_f32(Ap, EE, Bp, RR, EE);
  for (int i = 0; i < 8; ++i) {
    int row = (lane < 16) ? i : i + 8;
    int m = mt * 16 + row;
    int r = r0 + lo;
    hdn[(size_t)m * (6 * RR) + g * RR + r] = tanhf(c[i] + Ab[g * RR + r]);
  }
}

// ---------------- K2b: dd = x + diff * (lambda + hdn @ lora_B_w + b) ----------------
__global__ void k_loraB_dd(const float* __restrict__ hdn, const float* __restrict__ Bw,
                           const float* __restrict__ Bb, const float* __restrict__ lam,
                           const float* __restrict__ x, const float* __restrict__ diff,
                           float* __restrict__ dd) {
  int mt = blockIdx.x, nt = blockIdx.y, g = blockIdx.z;
  int lane = threadIdx.x & 31, lo = lane & 15;
  const float* Ap = hdn + (size_t)mt * 16 * (6 * RR) + g * RR;
  const float* Bp = Bw + (size_t)g * RR * EE + nt * 16;
  v8f c = wmma_tile_f32(Ap, 6 * RR, Bp, EE, RR);
  for (int i = 0; i < 8; ++i) {
    int row = (lane < 16) ? i : i + 8;
    int m = mt * 16 + row;
    int e = nt * 16 + lo;
    size_t xi = (size_t)m * EE + e;
    float lv = lam[g * EE + e] + c[i] + Bb[g * EE + e];
    dd[(size_t)m * (6 * EE) + g * EE + e] = x[xi] + diff[xi] * lv;
  }
}

// ---------------- K3: per-head dense D->D for branches 0..4 ----------------
__global__ void k_mhd(const float* __restrict__ dd, const float* __restrict__ W,
                      float* __restrict__ proj) {
  int mt = blockIdx.x;
  int h = blockIdx.y >> 2, nt = blockIdx.y & 3;
  int br = blockIdx.z;
  int lane = threadIdx.x & 31, lo = lane & 15;
  const float* Ap = dd + (size_t)mt * 16 * (6 * EE) + br * EE + h * DD;
  const float* Bp = W + (((size_t)br * HH + h) * DD) * DD + nt * 16;
  v8f c = wmma_tile_f32(Ap, 6 * EE, Bp, DD, DD);
  for (int i = 0; i < 8; ++i) {
    int row = (lane < 16) ? i : i + 8;
    size_t m = (size_t)mt * 16 + row;
    proj[((m * 5 + br) * HH + h) * DD + nt * 16 + lo] = c[i];
  }
}

// ---------------- K4: g = dd[:,:,5] @ gate_w  (2x2 blocked, 32x32 per wave) ----------------
__global__ void k_gate(const float* __restrict__ dd, const float* __restrict__ gw,
                       float* __restrict__ g) {
  int mt = blockIdx.x, nt = blockIdx.y;
  int lane = threadIdx.x & 31, lo = lane & 15;
  const float* Ap = dd + (size_t)mt * 32 * (6 * EE) + 5 * EE;
  const float* Bp = gw + nt * 32;
  v8f c[2][2] = {};
  wmma_tile_f32_2x2(Ap, 6 * EE, Bp, EE, EE, c);
  for (int mi = 0; mi < 2; ++mi)
    for (int nj = 0; nj < 2; ++nj)
      for (int i = 0; i < 8; ++i) {
        int row = ((lane < 16) ? i : i + 8) + mi * 16;
        g[((size_t)mt * 32 + row) * EE + nt * 32 + nj * 16 + lo] = c[mi][nj][i];
      }
}

// ---------------- K5: wdec = exp(-exp(lamb + tanh(w @ wkv_A) @ wkv_B)) ----------------
__global__ void k_wdec(const float* __restrict__ proj, const float* __restrict__ wA,
                       const float* __restrict__ wB, const float* __restrict__ lamb,
                       float* __restrict__ wdec) {
  __shared__ float wh[16 * DD];
  int mt = blockIdx.x, h = blockIdx.y;
  int wv = threadIdx.x >> 5;              // 4 waves, one 16-col subtile each
  int lane = threadIdx.x & 31, lo = lane & 15;
  // branch 3 (decay) of proj; token stride = 5*H*D
  const float* Ap = proj + (((size_t)mt * 80 + 3) * HH + h) * DD;
  const float* Bp = wA + ((size_t)h * DD) * DD + wv * 16;
  v8f c = wmma_tile_f32(Ap, 5 * HH * DD, Bp, DD, DD);
  for (int i = 0; i < 8; ++i) {
    int row = (lane < 16) ? i : i + 8;
    wh[row * DD + wv * 16 + lo] = tanhf(c[i]);
  }
  __syncthreads();
  const float* Bp2 = wB + ((size_t)h * DD) * DD + wv * 16;
  v8f c2 = wmma_tile_f32(wh, DD, Bp2, DD, DD);
  for (int i = 0; i < 8; ++i) {
    int row = (lane < 16) ? i : i + 8;
    int n = wv * 16 + lo;
    float w2 = lamb[h * DD + n] + c2[i];
    wdec[(((size_t)mt * 16 + row) * HH + h) * DD + n] = expf(-expf(w2));
  }
}

// ---------------- K6: sequential WKV scan, double-buffered LDS ----------------
__global__ void k_wkv(const float* __restrict__ proj, const float* __restrict__ wdec,
                      float* __restrict__ rwkv) {
  // buf[p][0]=k  [1]=v  [2]=r  [3]=u  [4]=w
  __shared__ float buf[2][5][DD];
  int bh = blockIdx.x;
  int b = bh / HH, h = bh % HH;
  int tid = threadIdx.x;
  int row = tid >> 2;            // state row d (0..63)
  int e0 = (tid & 3) * 16;       // 16 state columns per lane
  float S[16];
#pragma unroll
  for (int j = 0; j < 16; ++j) S[j] = 0.f;

  // prologue: stage token t=0 into buf[0]
  if (tid < DD) {
    size_t m0 = (size_t)b * TT;
    size_t base = ((m0 * 5) * HH + h) * DD + tid;
    buf[0][0][tid] = proj[base];
    buf[0][1][tid] = proj[base + (size_t)1 * HH * DD];
    buf[0][2][tid] = proj[base + (size_t)2 * HH * DD];
    buf[0][3][tid] = proj[base + (size_t)4 * HH * DD];
    buf[0][4][tid] = wdec[(m0 * HH + h) * DD + tid];
  }
  int p = 0;
  for (int t = 0; t < TT; ++t) {
    __syncthreads();
    // prefetch token t+1 into the other buffer while computing token t
    if (tid < DD && t + 1 < TT) {
      size_t m1 = (size_t)b * TT + t + 1;
      size_t base = ((m1 * 5) * HH + h) * DD + tid;
      buf[p ^ 1][0][tid] = proj[base];
      buf[p ^ 1][1][tid] = proj[base + (size_t)1 * HH * DD];
      buf[p ^ 1][2][tid] = proj[base + (size_t)2 * HH * DD];
      buf[p ^ 1][3][tid] = proj[base + (size_t)4 * HH * DD];
      buf[p ^ 1][4][tid] = wdec[(m1 * HH + h) * DD + tid];
    }
    float kd = buf[p][0][row], ud = buf[p][3][row];
    const float4* v4 = (const float4*)&buf[p][1][e0];
    const float4* r4 = (const float4*)&buf[p][2][e0];
    const float4* w4 = (const float4*)&buf[p][4][e0];
    float acc = 0.f;
#pragma unroll
    for (int j = 0; j < 4; ++j) {
      float4 vv = v4[j], rr = r4[j], ww = w4[j];
      float kv;
      kv = kd * vv.x; acc += (ud * kv + S[4*j+0]) * rr.x; S[4*j+0] = kv + ww.x * S[4*j+0];
      kv = kd * vv.y; acc += (ud * kv + S[4*j+1]) * rr.y; S[4*j+1] = kv + ww.y * S[4*j+1];
      kv = kd * vv.z; acc += (ud * kv + S[4*j+2]) * rr.z; S[4*j+2] = kv + ww.z * S[4*j+2];
      kv = kd * vv.w; acc += (ud * kv + S[4*j+3]) * rr.w; S[4*j+3] = kv + ww.w * S[4*j+3];
    }
    acc += __shfl_xor(acc, 1, 32);
    acc += __shfl_xor(acc, 2, 32);
    if ((tid & 3) == 0) {
      size_t m = (size_t)b * TT + t;
      rwkv[(m * HH + h) * DD + row] = acc;
    }
    p ^= 1;
  }
}

// ---------------- K7: GroupNorm(H groups) + silu(g) gate ----------------
__global__ void k_gn(const float* __restrict__ rwkv, const float* __restrict__ gamma,
                     const float* __restrict__ beta, const float* __restrict__ g,
                     float* __restrict__ y) {
  int idx = blockIdx.x;           // m*HH + h
  int lane = threadIdx.x;
  int h = idx & (HH - 1);
  int m = idx >> 4;
  size_t base = (size_t)idx * DD;
  float a0 = rwkv[base + lane], a1 = rwkv[base + lane + 32];
  float s = a0 + a1, sq = a0 * a0 + a1 * a1;
  for (int d = 1; d < 32; d <<= 1) { s += __shfl_xor(s, d, 32); sq += __shfl_xor(sq, d, 32); }
  float mean = s * (1.f / DD);
  float var  = sq * (1.f / DD) - mean * mean;
  float inv  = rsqrtf(var + 1e-3f);
  int e0 = h * DD + lane;
  size_t yb = (size_t)m * EE + h * DD;
  float n0 = (a0 - mean) * inv * gamma[e0] + beta[e0];
  float n1 = (a1 - mean) * inv * gamma[e0 + 32] + beta[e0 + 32];
  float g0 = g[yb + lane], g1 = g[yb + lane + 32];
  y[yb + lane]      = n0 * (g0 / (1.f + expf(-g0)));
  y[yb + lane + 32] = n1 * (g1 / (1.f + expf(-g1)));
}

// ---------------- K8: out = y @ out_w + out_b  (2x2 blocked) ----------------
__global__ void k_out(const float* __restrict__ y, const float* __restrict__ ow,
                      const float* __restrict__ ob, float* __restrict__ out) {
  int mt = blockIdx.x, nt = blockIdx.y;
  int lane = threadIdx.x & 31, lo = lane & 15;
  const float* Ap = y + (size_t)mt * 32 * EE;
  const float* Bp = ow + nt * 32;
  v8f c[2][2] = {};
  wmma_tile_f32_2x2(Ap, EE, Bp, EE, EE, c);
  for (int mi = 0; mi < 2; ++mi)
    for (int nj = 0; nj < 2; ++nj)
      for (int i = 0; i < 8; ++i) {
        int row = ((lane < 16) ? i : i + 8) + mi * 16;
        int n = nt * 32 + nj * 16 + lo;
        out[((size_t)mt * 32 + row) * EE + n] = c[mi][nj][i] + ob[n];
      }
}

extern "C" void kernel_launch(void* const* d_in, const int* in_sizes, int n_in,
                              void* d_out, int out_size, void* d_ws, size_t ws_size,
                              hipStream_t stream) {
  (void)in_sizes; (void)n_in; (void)out_size; (void)ws_size;
  const float* x    = (const float*)d_in[0];
  const float* mu_x = (const float*)d_in[1];
  const float* lam  = (const float*)d_in[2];
  const float* Aw   = (const float*)d_in[3];
  const float* Ab   = (const float*)d_in[4];
  const float* Bw   = (const float*)d_in[5];
  const float* Bb   = (const float*)d_in[6];
  const float* mhd  = (const float*)d_in[7];
  const float* gw   = (const float*)d_in[8];
  const float* wl   = (const float*)d_in[9];
  const float* wA   = (const float*)d_in[10];
  const float* wB   = (const float*)d_in[11];
  const float* gg   = (const float*)d_in[12];
  const float* gb   = (const float*)d_in[13];
  const float* ow   = (const float*)d_in[14];
  const float* ob   = (const float*)d_in[15];

  float* ws   = (float*)d_ws;
  float* z    = ws;                    // BT*E
  float* diff = ws + 2097152;          // BT*E
  float* hdn  = ws + 4194304;          // BT*192
  float* dd   = ws + 4718592;          // BT*6*E
  float* proj = ws + 17301504;         // BT*5*H*D
  float* wdec = ws + 27787264;         // BT*H*D   (end: 29,884,416 floats)
  float* gbuf = diff;                  // reuse: diff dead after K2b
  float* rwkv = dd;                    // reuse: dd dead after K4
  float* y    = z;                     // reuse: z dead after K2a

  k_prep<<<dim3((BT * EE) / 256), 256, 0, stream>>>(x, mu_x, z, diff);
  k_loraA<<<dim3(BT / 16, 12), 32, 0, stream>>>(z, Aw, Ab, hdn);
  k_loraB_dd<<<dim3(BT / 16, EE / 16, 6), 32, 0, stream>>>(hdn, Bw, Bb, lam, x, diff, dd);
  k_mhd<<<dim3(BT / 16, HH * 4, 5), 32, 0, stream>>>(dd, mhd, proj);
  k_gate<<<dim3(BT / 32, EE / 32), 32, 0, stream>>>(dd, gw, gbuf);
  k_wdec<<<dim3(BT / 16, HH), 128, 0, stream>>>(proj, wA, wB, wl, wdec);
  k_wkv<<<dim3(BB * HH), 256, 0, stream>>>(proj, wdec, rwkv);
  k_gn<<<dim3(BT * HH), 32, 0, stream>>>(rwkv, gg, gb, gbuf, y);
  k_out<<<dim3(BT / 32, EE / 32), 32, 0, stream>>>(y, ow, ob, (float*)d_out);
}